// mtrnn_75479755260396
// MI455X (gfx1250) — compile-verified
//
#include <hip/hip_runtime.h>
#include <math.h>

typedef float v2f __attribute__((ext_vector_type(2)));
typedef float v8f __attribute__((ext_vector_type(8)));

namespace {
constexpr int T_SEQ = 128;
constexpr int BATCH = 16384;
constexpr int NIN   = 28;     // input features
constexpr int NHID  = 100;    // hidden
constexpr int NCLS  = 10;     // classes
constexpr int HP    = 112;    // hidden padded to 7*16
constexpr int HS    = 113;    // LDS row stride for h (odd -> conflict free)
constexpr int NT    = 7;      // 16-wide N tiles over HP
constexpr int KX    = NIN / 4;   // 7 K-steps over input dim
constexpr int KH    = NHID / 4;  // 25 K-steps over hidden dim
constexpr int MTILE = 128;    // batch rows per workgroup
constexpr int BLOCK = 128;    // 4 waves; each wave owns 32 rows
}

__device__ __forceinline__ v8f wmma_f32(v2f a, v2f b, v8f c) {
  // D = A(16x4) * B(4x16) + C(16x16), full fp32
  return __builtin_amdgcn_wmma_f32_16x16x4_f32(false, a, false, b, (short)0, c,
                                               false, false);
}

// tanh(x) = 1 - 2/(exp(2x)+1) on the native trans pipe (v_exp_f32, v_rcp_f32).
// Exact saturation: exp->inf => 1-0 = +1 ; exp->0 => 1-2 = -1.
__device__ __forceinline__ float fast_tanhf(float x) {
  float e = __expf(2.0f * x);
  return 1.0f - 2.0f * __builtin_amdgcn_rcpf(e + 1.0f);
}

__global__ __launch_bounds__(BLOCK) void rnn_fused_kernel(
    const float* __restrict__ data, const float* __restrict__ w_xh,
    const float* __restrict__ b_xh, const float* __restrict__ w_hh,
    const float* __restrict__ b_hh, const float* __restrict__ w_fc,
    const float* __restrict__ b_fc, float* __restrict__ out) {
  __shared__ float s_x[2][MTILE * NIN];  // double-buffered x_t tile
  __shared__ float s_h[MTILE][HS];       // hidden state tile
  __shared__ float s_whh[HP][HP];        // W_hh  [k][n], zero padded
  __shared__ float s_wxh[NIN][HP];       // W_xh  [k][n], zero padded
  __shared__ float s_wfc[HP][16];        // W_fc  [k][n], zero padded
  __shared__ float s_bias[HP];           // b_xh + b_hh, zero padded
  __shared__ float s_bfc[16];

  const int tid   = threadIdx.x;
  const int lane  = tid & 31;
  const int wave  = tid >> 5;              // 0..3
  const int col   = lane & 15;             // N index within 16-tile (B/C/D)
  const int lrow  = lane & 15;             // M index within 16-tile (A)
  const int khalf = (lane >> 4) << 1;      // 0 or 2: K pair selector (A/B frags)
  const int rhalf = (lane >> 4) << 3;      // 0 or 8: M offset (C/D frags)
  const int rowA0 = wave * 32;             // first 16-row group of this wave
  const int rowA1 = rowA0 + 16;            // second 16-row group
  const int b0    = blockIdx.x * MTILE;

  // ---- one-time staging: zero h, pad+stage weights/biases ----
  for (int i = tid; i < MTILE * HS; i += BLOCK) (&s_h[0][0])[i] = 0.0f;
  for (int i = tid; i < HP * HP; i += BLOCK) {
    int k = i / HP, n = i % HP;
    s_whh[k][n] = (k < NHID && n < NHID) ? w_hh[k * NHID + n] : 0.0f;
  }
  for (int i = tid; i < NIN * HP; i += BLOCK) {
    int k = i / HP, n = i % HP;
    s_wxh[k][n] = (n < NHID) ? w_xh[k * NHID + n] : 0.0f;
  }
  for (int i = tid; i < HP * 16; i += BLOCK) {
    int k = i >> 4, n = i & 15;
    s_wfc[k][n] = (k < NHID && n < NCLS) ? w_fc[k * NCLS + n] : 0.0f;
  }
  for (int i = tid; i < HP; i += BLOCK)
    s_bias[i] = (i < NHID) ? (b_xh[i] + b_hh[i]) : 0.0f;
  if (tid < 16) s_bfc[tid] = (tid < NCLS) ? b_fc[tid] : 0.0f;
  __syncthreads();

  // bias accumulator fragments, hoisted out of the time loop
  v8f cbias[NT];
#pragma unroll
  for (int n = 0; n < NT; ++n) {
    const float bv = s_bias[n * 16 + col];
    v8f c;
#pragma unroll
    for (int r = 0; r < 8; ++r) c[r] = bv;
    cbias[n] = c;
  }

#pragma unroll 1
  for (int t = 0; t < T_SEQ; ++t) {
    // stage x_t into buffer (t&1): contiguous MTILE*NIN floats, coalesced
    float* xb = s_x[t & 1];
    const float4* xt =
        (const float4*)(data + ((size_t)t * BATCH + b0) * NIN);
    for (int i = tid; i < MTILE * NIN / 4; i += BLOCK)
      ((float4*)xb)[i] = xt[i];
    // L2 prefetch of next timestep's tile (one cacheline per thread)
    if (t + 1 < T_SEQ) {
      const float* xn = data + ((size_t)(t + 1) * BATCH + b0) * NIN;
      for (int i = tid * 32; i < MTILE * NIN; i += BLOCK * 32)
        __builtin_prefetch(xn + i, 0, 1);
    }
    __syncthreads();  // single barrier per step (double-buffered x)

    // accumulators preloaded with fused bias (b_xh + b_hh)
    v8f acc0[NT], acc1[NT];
#pragma unroll
    for (int n = 0; n < NT; ++n) {
      acc0[n] = cbias[n];
      acc1[n] = cbias[n];
    }

    // x_t @ W_xh : K = 28 (7 steps of 4)
#pragma unroll
    for (int k = 0; k < KX; ++k) {
      const int kb = k * 4 + khalf;
      v2f a0, a1;
      a0.x = xb[(rowA0 + lrow) * NIN + kb];
      a0.y = xb[(rowA0 + lrow) * NIN + kb + 1];
      a1.x = xb[(rowA1 + lrow) * NIN + kb];
      a1.y = xb[(rowA1 + lrow) * NIN + kb + 1];
#pragma unroll
      for (int n = 0; n < NT; ++n) {
        v2f b;
        b.x = s_wxh[kb][n * 16 + col];
        b.y = s_wxh[kb + 1][n * 16 + col];
        acc0[n] = wmma_f32(a0, b, acc0[n]);
        acc1[n] = wmma_f32(a1, b, acc1[n]);
      }
    }

    // h @ W_hh : K = 100 (25 steps of 4); wave reads only its own h rows
#pragma unroll 1
    for (int k = 0; k < KH; ++k) {
      const int kb = k * 4 + khalf;
      v2f a0, a1;
      a0.x = s_h[rowA0 + lrow][kb];
      a0.y = s_h[rowA0 + lrow][kb + 1];
      a1.x = s_h[rowA1 + lrow][kb];
      a1.y = s_h[rowA1 + lrow][kb + 1];
#pragma unroll
      for (int n = 0; n < NT; ++n) {
        v2f b;
        b.x = s_whh[kb][n * 16 + col];
        b.y = s_whh[kb + 1][n * 16 + col];
        acc0[n] = wmma_f32(a0, b, acc0[n]);
        acc1[n] = wmma_f32(a1, b, acc1[n]);
      }
    }

    // h = tanh(pre); write back own rows (no cross-wave sharing of h)
#pragma unroll
    for (int n = 0; n < NT; ++n) {
#pragma unroll
      for (int r = 0; r < 8; ++r) {
        s_h[rowA0 + rhalf + r][n * 16 + col] = fast_tanhf(acc0[n][r]);
        s_h[rowA1 + rhalf + r][n * 16 + col] = fast_tanhf(acc1[n][r]);
      }
    }
  }

  // ---- final FC: out = h_T @ W_fc + b_fc ----
  v8f f0, f1;
  {
    const float bv = s_bfc[col];
#pragma unroll
    for (int r = 0; r < 8; ++r) {
      f0[r] = bv;
      f1[r] = bv;
    }
  }
#pragma unroll 1
  for (int k = 0; k < KH; ++k) {
    const int kb = k * 4 + khalf;
    v2f a0, a1, b;
    a0.x = s_h[rowA0 + lrow][kb];
    a0.y = s_h[rowA0 + lrow][kb + 1];
    a1.x = s_h[rowA1 + lrow][kb];
    a1.y = s_h[rowA1 + lrow][kb + 1];
    b.x = s_wfc[kb][col];
    b.y = s_wfc[kb + 1][col];
    f0 = wmma_f32(a0, b, f0);
    f1 = wmma_f32(a1, b, f1);
  }
  if (col < NCLS) {
#pragma unroll
    for (int r = 0; r < 8; ++r) {
      out[(size_t)(b0 + rowA0 + rhalf + r) * NCLS + col] = f0[r];
      out[(size_t)(b0 + rowA1 + rhalf + r) * NCLS + col] = f1[r];
    }
  }
}

extern "C" void kernel_launch(void* const* d_in, const int* in_sizes, int n_in,
                              void* d_out, int out_size, void* d_ws,
                              size_t ws_size, hipStream_t stream) {
  const float* data = (const float*)d_in[0];
  const float* w_xh = (const float*)d_in[1];
  const float* b_xh = (const float*)d_in[2];
  const float* w_hh = (const float*)d_in[3];
  const float* b_hh = (const float*)d_in[4];
  const float* w_fc = (const float*)d_in[5];
  const float* b_fc = (const float*)d_in[6];
  float* out = (float*)d_out;

  dim3 grid(BATCH / MTILE);  // 128 workgroups
  dim3 block(BLOCK);         // 4 waves of 32
  hipLaunchKernelGGL(rnn_fused_kernel, grid, block, 0, stream, data, w_xh,
                     b_xh, w_hh, b_hh, w_fc, b_fc, out);
}